// GNN_11957188952096
// MI455X (gfx1250) — compile-verified
//
#include <hip/hip_runtime.h>
#include <hip/hip_bf16.h>
#include <cstddef>

typedef __attribute__((ext_vector_type(2))) float v2f;
typedef __attribute__((ext_vector_type(8))) float v8f;

#define EPSN 1e-12f

// ---------------------------------------------------------------------------
// pre: h0 = relu(x @ pre_w + pre_b), x:[n,5], w:[5,32]
// ---------------------------------------------------------------------------
__global__ void pre_kernel(const float* __restrict__ x, const float* __restrict__ w,
                           const float* __restrict__ b, float* __restrict__ h0, int n) {
    int node = blockIdx.x * blockDim.x + threadIdx.x;
    if (node >= n) return;
    float xv[5];
#pragma unroll
    for (int k = 0; k < 5; ++k) xv[k] = x[(size_t)node * 5 + k];
#pragma unroll
    for (int o = 0; o < 32; ++o) {
        float a = b[o];
#pragma unroll
        for (int k = 0; k < 5; ++k) a = fmaf(xv[k], w[k * 32 + o], a);
        h0[(size_t)node * 32 + o] = fmaxf(a, 0.0f);
    }
}

// ---------------------------------------------------------------------------
// degree count + inverse count
// ---------------------------------------------------------------------------
__global__ void count_edges_kernel(const int* __restrict__ dst, int e, int* __restrict__ cnt) {
    int i = blockIdx.x * blockDim.x + threadIdx.x;
    if (i < e) atomicAdd(&cnt[dst[i]], 1);
}

__global__ void invcnt_kernel(const int* __restrict__ cnt, float* __restrict__ invc, int n) {
    int i = blockIdx.x * blockDim.x + threadIdx.x;
    if (i < n) {
        int c = cnt[i];
        invc[i] = (c > 0) ? (1.0f / (float)c) : 0.0f;
    }
}

// ---------------------------------------------------------------------------
// scatter: agg[dst] += h[src]  (D = feature dim, thread per (edge, 4-feature))
// ---------------------------------------------------------------------------
template <int D>
__global__ void scatter_kernel(const float* __restrict__ h, const int* __restrict__ src,
                               const int* __restrict__ dst, float* __restrict__ agg, int e) {
    const int per = D / 4;
    int idx = blockIdx.x * blockDim.x + threadIdx.x;
    int eidx = idx / per;
    int f = (idx - eidx * per) * 4;
    if (eidx >= e) return;
    int s = src[eidx];
    int d = dst[eidx];
    const float4 v = *(const float4*)(h + (size_t)s * D + f);
    float* p = agg + (size_t)d * D + f;
    unsafeAtomicAdd(p + 0, v.x);
    unsafeAtomicAdd(p + 1, v.y);
    unsafeAtomicAdd(p + 2, v.z);
    unsafeAtomicAdd(p + 3, v.w);
}

// ---------------------------------------------------------------------------
// SAGEConv dense part with fp32 WMMA (16x16x4):
//   out = relu( l2norm( (agg*invcnt) @ Wl + bl + h @ Wr ) )
// One wave = one 16-node tile, 4 output column tiles of 16 (out dim 64).
//
// f32 WMMA fragment layouts (ISA 7.12.2):
//   A (16x4):  lane m=L&15 -> row M=m; koff=(L>>4)*2; vgpr0=K=koff, vgpr1=K=koff+1
//   B (4x16):  lane m=L&15 -> col N=m; vgpr0=row K=koff, vgpr1=row K=koff+1
//   C/D:       lane -> col N=L&15; vgpr j -> row M=j+8*(L>>4)
//
// Weights are staged in LDS in *fragment-pair order*: pair index (k>>1)*64+n
// holds (W[k][n], W[k+1][n]) so every B fragment is one aligned ds_load_b64.
// ---------------------------------------------------------------------------
template <int K>
__global__ __launch_bounds__(256) void sage_wmma_kernel(
    const float* __restrict__ agg,    // [n, K] neighbor sums
    const float* __restrict__ invc,   // [n]
    const float* __restrict__ h,      // [n, K] root features
    const float* __restrict__ wl,     // [K, 64]
    const float* __restrict__ bl,     // [64]
    const float* __restrict__ wr,     // [K, 64]
    float* __restrict__ out,          // [n, 64]
    int n) {
    __shared__ v2f s_wl[(K / 2) * 64];
    __shared__ v2f s_wr[(K / 2) * 64];
    __shared__ float s_bl[64];
    {
        float* fwl = (float*)s_wl;
        float* fwr = (float*)s_wr;
        for (int i = threadIdx.x; i < K * 64; i += 256) {
            int k = i >> 6;
            int nn = i & 63;
            int sidx = ((((k >> 1) * 64) + nn) << 1) | (k & 1);
            fwl[sidx] = wl[i];
            fwr[sidx] = wr[i];
        }
    }
    if (threadIdx.x < 64) s_bl[threadIdx.x] = bl[threadIdx.x];
    __syncthreads();

    const int lane = threadIdx.x & 31;
    const int wave = threadIdx.x >> 5;
    const int base = (blockIdx.x * 8 + wave) * 16;
    if (base >= n) return;

    const int m = lane & 15;
    const int koff = (lane >> 4) << 1;  // 0 or 2
    int row = base + m;
    if (row >= n) row = n - 1;  // clamp for partial tile (stores are guarded)
    const float inv = invc[row];
    const float* aggr = agg + (size_t)row * K;
    const float* hr = h + (size_t)row * K;

    v8f acc[4] = {v8f{}, v8f{}, v8f{}, v8f{}};

    for (int kk = 0; kk < K; kk += 4) {
        v2f a_agg, a_h;
        a_agg.x = aggr[kk + koff] * inv;
        a_agg.y = aggr[kk + koff + 1] * inv;
        a_h.x = hr[kk + koff];
        a_h.y = hr[kk + koff + 1];
        const int prow = ((kk + koff) >> 1) * 64 + m;  // fragment-pair row base
#pragma unroll
        for (int t = 0; t < 4; ++t) {
            v2f bL = s_wl[prow + t * 16];
            acc[t] = __builtin_amdgcn_wmma_f32_16x16x4_f32(
                false, a_agg, false, bL, (short)0, acc[t], false, false);
            v2f bR = s_wr[prow + t * 16];
            acc[t] = __builtin_amdgcn_wmma_f32_16x16x4_f32(
                false, a_h, false, bR, (short)0, acc[t], false, false);
        }
    }

    float bias_t[4];
#pragma unroll
    for (int t = 0; t < 4; ++t) bias_t[t] = s_bl[t * 16 + m];

#pragma unroll
    for (int j = 0; j < 8; ++j) {
        float v0 = acc[0][j] + bias_t[0];
        float v1 = acc[1][j] + bias_t[1];
        float v2 = acc[2][j] + bias_t[2];
        float v3 = acc[3][j] + bias_t[3];
        float s = v0 * v0 + v1 * v1 + v2 * v2 + v3 * v3;
        // reduce across the 16 lanes that hold this row's 64 columns
        s += __shfl_xor(s, 1, 16);
        s += __shfl_xor(s, 2, 16);
        s += __shfl_xor(s, 4, 16);
        s += __shfl_xor(s, 8, 16);
        float rn = 1.0f / fmaxf(sqrtf(s), EPSN);
        int r = base + j + ((lane >> 4) << 3);
        if (r < n) {
            float* o = out + (size_t)r * 64 + m;
            o[0] = fmaxf(v0 * rn, 0.0f);
            o[16] = fmaxf(v1 * rn, 0.0f);
            o[32] = fmaxf(v2 * rn, 0.0f);
            o[48] = fmaxf(v3 * rn, 0.0f);
        }
    }
}

// ---------------------------------------------------------------------------
// global mean pool accumulate: gsum[batch[n]] += h2[n]; gcnt[batch[n]] += 1
// ---------------------------------------------------------------------------
__global__ void pool_kernel(const float* __restrict__ h2, const int* __restrict__ batch,
                            float* __restrict__ gsum, int* __restrict__ gcnt, int n) {
    int idx = blockIdx.x * blockDim.x + threadIdx.x;
    int node = idx >> 4;
    int f = (idx & 15) * 4;
    if (node >= n) return;
    int b = batch[node];
    float4 v = *(const float4*)(h2 + (size_t)node * 64 + f);
    float* p = gsum + (size_t)b * 64 + f;
    unsafeAtomicAdd(p + 0, v.x);
    unsafeAtomicAdd(p + 1, v.y);
    unsafeAtomicAdd(p + 2, v.z);
    unsafeAtomicAdd(p + 3, v.w);
    if (f == 0) atomicAdd(&gcnt[b], 1);
}

// ---------------------------------------------------------------------------
// head MLP: g -> relu(g@p1+b1) -> relu(@p2+b2) -> @o + ob ; block per graph
// ---------------------------------------------------------------------------
__global__ __launch_bounds__(64) void mlp_kernel(
    const float* __restrict__ gsum, const int* __restrict__ gcnt,
    const float* __restrict__ p1w, const float* __restrict__ p1b,
    const float* __restrict__ p2w, const float* __restrict__ p2b,
    const float* __restrict__ ow, const float* __restrict__ ob,
    float* __restrict__ out, int ngraphs) {
    int g = blockIdx.x;
    if (g >= ngraphs) return;
    __shared__ float sg[64];
    __shared__ float s1[64];
    __shared__ float s2[16];
    int t = threadIdx.x;
    float inv = 1.0f / fmaxf((float)gcnt[g], 1.0f);
    sg[t] = gsum[(size_t)g * 64 + t] * inv;
    __syncthreads();
    float a = p1b[t];
    for (int k = 0; k < 64; ++k) a = fmaf(sg[k], p1w[k * 64 + t], a);
    s1[t] = fmaxf(a, 0.0f);
    __syncthreads();
    if (t < 16) {
        float b = p2b[t];
        for (int k = 0; k < 64; ++k) b = fmaf(s1[k], p2w[k * 16 + t], b);
        s2[t] = fmaxf(b, 0.0f);
    }
    __syncthreads();
    if (t == 0) {
        float o = ob[0];
        for (int k = 0; k < 16; ++k) o = fmaf(s2[k], ow[k], o);
        out[g] = o;
    }
}

// ---------------------------------------------------------------------------
extern "C" void kernel_launch(void* const* d_in, const int* in_sizes, int n_in,
                              void* d_out, int out_size, void* d_ws, size_t ws_size,
                              hipStream_t stream) {
    const float* x = (const float*)d_in[0];
    const int* eidx = (const int*)d_in[1];
    const int* batch = (const int*)d_in[2];
    const float* pre_w = (const float*)d_in[4];
    const float* pre_b = (const float*)d_in[5];
    const float* c1_wl = (const float*)d_in[6];
    const float* c1_bl = (const float*)d_in[7];
    const float* c1_wr = (const float*)d_in[8];
    const float* c2_wl = (const float*)d_in[9];
    const float* c2_bl = (const float*)d_in[10];
    const float* c2_wr = (const float*)d_in[11];
    const float* p1_w = (const float*)d_in[12];
    const float* p1_b = (const float*)d_in[13];
    const float* p2_w = (const float*)d_in[14];
    const float* p2_b = (const float*)d_in[15];
    const float* o_w = (const float*)d_in[16];
    const float* o_b = (const float*)d_in[17];
    float* out = (float*)d_out;

    const int n = in_sizes[0] / 5;
    const int e = in_sizes[1] / 2;
    const int G = out_size;
    const int* src = eidx;
    const int* dstv = eidx + e;

    float* ws = (float*)d_ws;
    size_t off = 0;
    float* h0 = ws + off;   off += (size_t)n * 32;
    float* h1 = ws + off;   off += (size_t)n * 64;
    float* h2 = ws + off;   off += (size_t)n * 64;
    float* agg = ws + off;  off += (size_t)n * 64;
    int* cnt = (int*)(ws + off);   off += n;
    float* invc = ws + off;        off += n;
    float* gsum = ws + off;        off += (size_t)G * 64;
    int* gcnt = (int*)(ws + off);  off += G;

    const int TB = 256;

    // degree counts (shared by both convs)
    hipMemsetAsync(cnt, 0, (size_t)n * sizeof(int), stream);
    count_edges_kernel<<<(e + TB - 1) / TB, TB, 0, stream>>>(dstv, e, cnt);
    invcnt_kernel<<<(n + TB - 1) / TB, TB, 0, stream>>>(cnt, invc, n);

    // pre linear
    pre_kernel<<<(n + TB - 1) / TB, TB, 0, stream>>>(x, pre_w, pre_b, h0, n);

    // conv1 (K=32)
    hipMemsetAsync(agg, 0, (size_t)n * 32 * sizeof(float), stream);
    {
        long long thr = (long long)e * 8;
        scatter_kernel<32><<<(unsigned)((thr + TB - 1) / TB), TB, 0, stream>>>(h0, src, dstv, agg, e);
    }
    {
        int tiles = (n + 15) / 16;
        int blocks = (tiles + 7) / 8;
        sage_wmma_kernel<32><<<blocks, TB, 0, stream>>>(agg, invc, h0, c1_wl, c1_bl, c1_wr, h1, n);
    }

    // conv2 (K=64)
    hipMemsetAsync(agg, 0, (size_t)n * 64 * sizeof(float), stream);
    {
        long long thr = (long long)e * 16;
        scatter_kernel<64><<<(unsigned)((thr + TB - 1) / TB), TB, 0, stream>>>(h1, src, dstv, agg, e);
    }
    {
        int tiles = (n + 15) / 16;
        int blocks = (tiles + 7) / 8;
        sage_wmma_kernel<64><<<blocks, TB, 0, stream>>>(agg, invc, h1, c2_wl, c2_bl, c2_wr, h2, n);
    }

    // global mean pool
    hipMemsetAsync(gsum, 0, (size_t)G * 64 * sizeof(float), stream);
    hipMemsetAsync(gcnt, 0, (size_t)G * sizeof(int), stream);
    {
        long long thr = (long long)n * 16;
        pool_kernel<<<(unsigned)((thr + TB - 1) / TB), TB, 0, stream>>>(h2, batch, gsum, gcnt, n);
    }

    // head MLP -> out
    mlp_kernel<<<G, 64, 0, stream>>>(gsum, gcnt, p1_w, p1_b, p2_w, p2_b, o_w, o_b, out, G);
}